// HarmonicLayer_30605936951521
// MI455X (gfx1250) — compile-verified
//
#include <hip/hip_runtime.h>
#include <hip/hip_bf16.h>
#include <math.h>

// ---------------------------------------------------------------------------
// Harmonic spectral layer on MI455X (gfx1250, wave32, WMMA).
// Heavy stages: batched fp32 complex GEMMs on v_wmma_f32_16x16x4_f32.
// LDS staging: global_load_async_to_lds_b128 (ASYNCcnt) — CDNA5 async path.
// ---------------------------------------------------------------------------

typedef float v2f __attribute__((ext_vector_type(2)));
typedef float v8f __attribute__((ext_vector_type(8)));

#define BC    128        // B*Cin = B*Cout
#define NN    64         // grid size per axis
#define MM    32         // retained modes per axis
#define VBLK  32768      // MM^3
#define NSH   16         // radial shells
#define NL    9          // spherical harmonics
#define NSL   144        // NSH*NL

// 16x16x4 fp32 WMMA: D = A(16x4) * B(4x16) + C(16x16)
__device__ __forceinline__ v8f wmma4(v2f a, v2f b, v8f c) {
  return __builtin_amdgcn_wmma_f32_16x16x4_f32(
      /*neg_a*/false, a, /*neg_b*/false, b, /*c_mod*/(short)0, c,
      /*reuse_a*/false, /*reuse_b*/false);
}

// Fragment load for both operands.
//  A operand: source stored row-major [m][k]  (16 rows, K-slab of 4)
//  B operand: source stored row-major [n][k]  (i.e. B^T), same access pattern
//  lane mapping: row = lane&15, k = 2*(lane>>4) + reg
__device__ __forceinline__ v2f load_frag(const float* __restrict__ src, int ld) {
  int lane = threadIdx.x & 31;
  const float* p = src + (lane & 15) * ld + 2 * (lane >> 4);
  v2f f; f.x = p[0]; f.y = p[1];
  return f;
}

#define V8F_ZERO {0.f,0.f,0.f,0.f,0.f,0.f,0.f,0.f}

// ---- CDNA5 async global->LDS copy (16B per lane per op, ASYNCcnt-tracked) --
__device__ __forceinline__ void async_ld_b128(float* lds_dst, const float* gsrc) {
  unsigned loff = (unsigned)(size_t)(void*)lds_dst;   // low 32b = LDS offset
  asm volatile("global_load_async_to_lds_b128 %0, %1, off"
               :: "v"(loff), "v"(gsrc) : "memory");
}
__device__ __forceinline__ void async_copy(float* lds_dst, const float* gsrc,
                                           int nfloats, int tid) {
  for (int i = tid * 4; i < nfloats; i += 256 * 4)
    async_ld_b128(lds_dst + i, gsrc + i);
}
__device__ __forceinline__ void async_wait() {
  asm volatile("s_wait_asynccnt 0x0" ::: "memory");
}

// ---------------------------------------------------------------------------
// Table init: DFT matrices.
// F[j][z] = exp(-2*pi*i*(j-16)*z/64)                 (forward, 32x64)
// G[z][j] = (1/64)*exp(+2*pi*i*(j-16)*z/64)          (inverse, 64x32, norm in)
// "*n" variants hold negated imaginary parts (f32 WMMA cannot negate A/B).
// ---------------------------------------------------------------------------
__global__ void init_dft_kernel(float* Fr, float* Fi, float* Fin,
                                float* Gr, float* Gi, float* Gin) {
  int tid = blockIdx.x * blockDim.x + threadIdx.x;
  if (tid >= MM * NN) return;
  const float TWO_PI = 6.283185307179586f;
  {
    int j = tid >> 6, z = tid & 63;                 // F: [32][64]
    float th = -TWO_PI * (float)((j - 16) * z) / 64.f;
    Fr[tid] = cosf(th);
    Fi[tid] = sinf(th);
    Fin[tid] = -sinf(th);
  }
  {
    int z = tid >> 5, j = tid & 31;                 // G: [64][32]
    float th = TWO_PI * (float)((j - 16) * z) / 64.f;
    const float s = 1.f / 64.f;
    Gr[tid] = s * cosf(th);
    Gi[tid] = s * sinf(th);
    Gin[tid] = -s * sinf(th);
  }
}

// ---------------------------------------------------------------------------
// Shell index + real spherical harmonics over the centered 32^3 block.
// ---------------------------------------------------------------------------
__global__ void init_basis_kernel(float* __restrict__ Y, int* __restrict__ shell,
                                  int* __restrict__ counts) {
  int v = blockIdx.x * 256 + threadIdx.x;           // grid covers 32768 exactly
  int ix = v >> 10, iy = (v >> 5) & 31, iz = v & 31;
  float x = (float)(ix - 16), y = (float)(iy - 16), z = (float)(iz - 16);
  float r = sqrtf(x * x + y * y + z * z);
  float rs = (r == 0.f) ? 1.f : r;
  float ux = x / rs, uy = y / rs, uz = z / rs;
  float Yv[NL];
  Yv[0] = 0.282095f;
  Yv[1] = 0.488603f * uy;  Yv[2] = 0.488603f * uz;  Yv[3] = 0.488603f * ux;
  Yv[4] = 1.092548f * ux * uy;
  Yv[5] = 1.092548f * uy * uz;
  Yv[6] = 0.315392f * (3.f * uz * uz - 1.f);
  Yv[7] = 1.092548f * ux * uz;
  Yv[8] = 0.546274f * (ux * ux - uy * uy);
  if (r == 0.f) {
#pragma unroll
    for (int l = 1; l < NL; ++l) Yv[l] = 0.f;
  }
  const float rmax = 27.712812921102035f;           // sqrt(3)*16
  int s = (int)(r / rmax * (float)NSH);
  if (s > NSH - 1) s = NSH - 1;
  shell[v] = s;
  atomicAdd(&counts[s], 1);
#pragma unroll
  for (int l = 0; l < NL; ++l) Y[l * VBLK + v] = Yv[l];
}

__global__ void init_counts_kernel(const int* __restrict__ counts,
                                   float* __restrict__ invcnt) {
  int t = threadIdx.x;
  if (t < NSH) invcnt[t] = 1.f / fmaxf((float)counts[t], 1.f);
}

// ---------------------------------------------------------------------------
// Forward, axes W then H fused per (b,c,d) slab.
//   T[h][j1]    = sum_w x[h][w]  * F[j1][w]          (real  x complex)
//   out[j2][j1] = sum_h F[j2][h] * T[h][j1]          (cplx  x cplx)
// Output stored as s2[bc][n=j2*32+j1][d]  (B^T layout for next GEMM stage).
// ---------------------------------------------------------------------------
__global__ __launch_bounds__(256) void fwd_stage12_kernel(
    const float* __restrict__ x,
    const float* __restrict__ Fr, const float* __restrict__ Fi,
    const float* __restrict__ Fin,
    float* __restrict__ s2r, float* __restrict__ s2i) {
  __shared__ float xs[NN * NN];                     // 16 KB
  __shared__ float fr[MM * NN], fi[MM * NN], fn[MM * NN];  // 24 KB
  __shared__ float ttr[MM * NN], tti[MM * NN];      // T^T [j1][h], 16 KB
  int slab = blockIdx.x;                            // bc*64 + d
  int tid = threadIdx.x, lane = tid & 31;
  int wave = __builtin_amdgcn_readfirstlane(tid >> 5);   // scalar wave id

  async_copy(xs, x + (size_t)slab * (NN * NN), NN * NN, tid);
  async_copy(fr, Fr, MM * NN, tid);
  async_copy(fi, Fi, MM * NN, tid);
  async_copy(fn, Fin, MM * NN, tid);
  async_wait();
  __syncthreads();

  // ---- stage A: 4 h-tiles x 2 j1-tiles x {r,i} = 16 wave-jobs (scalar loop)
  for (int job = wave; job < 16; job += 8) {
    int ri = job & 1, nt = (job >> 1) & 1, mt = job >> 2;
    const float* fsel = ri ? fi : fr;
    v8f acc = V8F_ZERO;
    for (int k0 = 0; k0 < NN; k0 += 4) {
      v2f a = load_frag(xs + (mt * 16) * NN + k0, NN);     // A[m=h][k=w]
      v2f b = load_frag(fsel + (nt * 16) * NN + k0, NN);   // Bsrc[n=j1][k=w]
      acc = wmma4(a, b, acc);
    }
    float* tsel = ri ? tti : ttr;
    int n = nt * 16 + (lane & 15);
    int mb = mt * 16 + 8 * (lane >> 4);
#pragma unroll
    for (int r = 0; r < 8; ++r) tsel[n * NN + mb + r] = acc[r];  // store T^T
  }
  __syncthreads();

  // ---- stage B: 2 j2-tiles x 2 j1-tiles x {r,i} = 8 wave-jobs
  {
    int job = wave;
    int ri = job & 1, nt = (job >> 1) & 1, mt = job >> 2;
    v8f acc = V8F_ZERO;
    for (int k0 = 0; k0 < NN; k0 += 4) {
      // out_r = Fr*Tr + (-Fi)*Ti ; out_i = Fr*Ti + Fi*Tr
      v2f a1 = load_frag(fr + (mt * 16) * NN + k0, NN);
      v2f b1 = load_frag((ri ? tti : ttr) + (nt * 16) * NN + k0, NN);
      acc = wmma4(a1, b1, acc);
      v2f a2 = load_frag((ri ? fi : fn) + (mt * 16) * NN + k0, NN);
      v2f b2 = load_frag((ri ? ttr : tti) + (nt * 16) * NN + k0, NN);
      acc = wmma4(a2, b2, acc);
    }
    int bc = slab >> 6, d = slab & 63;
    float* dst = (ri ? s2i : s2r) + (size_t)bc * (1024 * NN) + d;
    int j1 = nt * 16 + (lane & 15);
    int j2b = mt * 16 + 8 * (lane >> 4);
#pragma unroll
    for (int r = 0; r < 8; ++r)
      dst[(size_t)((j2b + r) * MM + j1) * NN] = acc[r];
  }
}

// ---------------------------------------------------------------------------
// Generic batched complex GEMM (per-bc), C[bc][m][n] = sum_k A[m][k]*B[bc][n][k]
//   forward D-axis: A = F (M=32, K=64), B = s2,     C = Xb   (nswap=0)
//   inverse D-axis: A = G (M=64, K=32), B = Xrec^T, C = t1   (nswap=1:
//     n=(j2*32+j1) stored as j1*32+j2 so the consumer can async-copy slabs)
// grid = (16 n-chunks of 64, 128 bc)
// ---------------------------------------------------------------------------
__global__ __launch_bounds__(256) void cgemm_stage_kernel(
    const float* __restrict__ Br, const float* __restrict__ Bi,
    const float* __restrict__ Ar, const float* __restrict__ Ai,
    const float* __restrict__ An,
    float* __restrict__ Cr, float* __restrict__ Ci,
    int M, int K, int nswap) {
  __shared__ float ar[2048], ai[2048], an[2048];    // 24 KB (M*K <= 2048)
  __shared__ float bsr[64 * 64], bsi[64 * 64];      // 32 KB (64 x K)
  int bc = blockIdx.y, nbase = blockIdx.x * 64;
  int tid = threadIdx.x, lane = tid & 31;
  int wave = __builtin_amdgcn_readfirstlane(tid >> 5);

  async_copy(ar, Ar, M * K, tid);
  async_copy(ai, Ai, M * K, tid);
  async_copy(an, An, M * K, tid);
  size_t bbase = (size_t)bc * 1024 * K + (size_t)nbase * K;
  async_copy(bsr, Br + bbase, 64 * K, tid);
  async_copy(bsi, Bi + bbase, 64 * K, tid);
  async_wait();
  __syncthreads();

  int njobs = (M >> 4) * 4 * 2;                     // Mtiles x 4 ntiles x {r,i}
  for (int job = wave; job < njobs; job += 8) {
    int ri = job & 1, nt = (job >> 1) & 3, mt = job >> 3;
    v8f acc = V8F_ZERO;
    for (int k0 = 0; k0 < K; k0 += 4) {
      // C_r = Ar*Br + (-Ai)*Bi ; C_i = Ar*Bi + Ai*Br
      v2f a1 = load_frag(ar + (mt * 16) * K + k0, K);
      v2f b1 = load_frag((ri ? bsi : bsr) + (nt * 16) * K + k0, K);
      acc = wmma4(a1, b1, acc);
      v2f a2 = load_frag((ri ? ai : an) + (mt * 16) * K + k0, K);
      v2f b2 = load_frag((ri ? bsr : bsi) + (nt * 16) * K + k0, K);
      acc = wmma4(a2, b2, acc);
    }
    float* dst = (ri ? Ci : Cr) + (size_t)bc * M * 1024;
    int n = nbase + nt * 16 + (lane & 15);
    int nst = nswap ? ((n & 31) * 32 + (n >> 5)) : n;
    int mb = mt * 16 + 8 * (lane >> 4);
#pragma unroll
    for (int r = 0; r < 8; ++r) dst[(size_t)(mb + r) * 1024 + nst] = acc[r];
  }
}

// ---------------------------------------------------------------------------
// Projection onto (shell x Ylm): LDS float atomics (ds_add_f32), then /counts.
// ---------------------------------------------------------------------------
__global__ __launch_bounds__(256) void project_kernel(
    const float* __restrict__ Xbr, const float* __restrict__ Xbi,
    const float* __restrict__ Y, const int* __restrict__ shell,
    const float* __restrict__ invcnt,
    float* __restrict__ cfr, float* __restrict__ cfi) {
  __shared__ float acc[NSL * 2];
  int bc = blockIdx.x, tid = threadIdx.x;
  for (int i = tid; i < NSL * 2; i += 256) acc[i] = 0.f;
  __syncthreads();
  const float* xr = Xbr + (size_t)bc * VBLK;
  const float* xi = Xbi + (size_t)bc * VBLK;
  for (int v = tid; v < VBLK; v += 256) {
    int s = shell[v];
    float re = xr[v], im = xi[v];
#pragma unroll
    for (int l = 0; l < NL; ++l) {
      float y = Y[l * VBLK + v];
      atomicAdd(&acc[(s * NL + l) * 2 + 0], re * y);
      atomicAdd(&acc[(s * NL + l) * 2 + 1], im * y);
    }
  }
  __syncthreads();
  if (tid < NSL) {
    float ic = invcnt[tid / NL];
    cfr[(size_t)bc * NSL + tid] = acc[tid * 2 + 0] * ic;
    cfi[(size_t)bc * NSL + tid] = acc[tid * 2 + 1] * ic;
  }
}

// ---------------------------------------------------------------------------
// Per-(s,l) complex channel mixing (tiny -> plain VALU).
// ---------------------------------------------------------------------------
__global__ void mix_kernel(const float* __restrict__ cfr, const float* __restrict__ cfi,
                           const float* __restrict__ Wr, const float* __restrict__ Wi,
                           float* __restrict__ c2r, float* __restrict__ c2i) {
  int idx = blockIdx.x * blockDim.x + threadIdx.x;  // (b,o,sl)
  if (idx >= 4 * 32 * NSL) return;
  int sl = idx % NSL;
  int bo = idx / NSL;
  int o = bo & 31, b = bo >> 5;
  float ar = 0.f, ai = 0.f;
  for (int i = 0; i < 32; ++i) {
    float cr = cfr[(size_t)(b * 32 + i) * NSL + sl];
    float ci = cfi[(size_t)(b * 32 + i) * NSL + sl];
    float wr = Wr[(size_t)(i * 32 + o) * NSL + sl];
    float wi = Wi[(size_t)(i * 32 + o) * NSL + sl];
    ar += cr * wr - ci * wi;
    ai += cr * wi + ci * wr;
  }
  c2r[idx] = ar;
  c2i[idx] = ai;
}

// ---------------------------------------------------------------------------
// Inverse projection: Xrec[v] = sum_l cf2[s(v),l] * Y[l,v]; written as
// Xrec[bc][n=(j2*32+j1)][j3]  (B^T layout for inverse D-axis GEMM).
// ---------------------------------------------------------------------------
__global__ __launch_bounds__(256) void unproject_kernel(
    const float* __restrict__ c2r, const float* __restrict__ c2i,
    const float* __restrict__ Y, const int* __restrict__ shell,
    float* __restrict__ Xr, float* __restrict__ Xi) {
  __shared__ float cr[NSL], ci[NSL];
  int bc = blockIdx.y, tid = threadIdx.x;
  if (tid < NSL) { cr[tid] = c2r[(size_t)bc * NSL + tid]; ci[tid] = c2i[(size_t)bc * NSL + tid]; }
  __syncthreads();
  int v = blockIdx.x * 256 + tid;
  int s = shell[v];
  float ar = 0.f, ai = 0.f;
#pragma unroll
  for (int l = 0; l < NL; ++l) {
    float y = Y[l * VBLK + v];
    ar += cr[s * NL + l] * y;
    ai += ci[s * NL + l] * y;
  }
  int j3 = v >> 10, n = v & 1023;
  size_t off = (size_t)bc * VBLK + (size_t)n * MM + j3;
  Xr[off] = ar;
  Xi[off] = ai;
}

// ---------------------------------------------------------------------------
// Inverse, axes H then W fused per (b,o,d) slab.
//   U[h][j1] = sum_j2 G[h][j2] * t1[j2][j1]
//   out[h][w]= sum_j1 U[h][j1] * G[w][j1]     (G row-major is exactly B^T here)
// t1 slabs arrive pre-transposed ([j1][j2]) thanks to cgemm nswap=1, so the
// LDS fill is a straight async b128 copy. Writes interleaved complex64.
// ---------------------------------------------------------------------------
__global__ __launch_bounds__(256) void inv_stage23_kernel(
    const float* __restrict__ t1r, const float* __restrict__ t1i,
    const float* __restrict__ Gr, const float* __restrict__ Gi,
    const float* __restrict__ Gn,
    float* __restrict__ out) {
  __shared__ float t1tr[MM * MM], t1ti[MM * MM];    // t1^T [j1][j2], 8 KB
  __shared__ float gr[NN * MM], gi[NN * MM], gn[NN * MM];  // 24 KB
  __shared__ float ur[NN * MM], ui[NN * MM];        // U [h][j1], 16 KB
  int slab = blockIdx.x;                            // bc*64 + d
  int tid = threadIdx.x, lane = tid & 31;
  int wave = __builtin_amdgcn_readfirstlane(tid >> 5);

  async_copy(t1tr, t1r + (size_t)slab * 1024, 1024, tid);
  async_copy(t1ti, t1i + (size_t)slab * 1024, 1024, tid);
  async_copy(gr, Gr, NN * MM, tid);
  async_copy(gi, Gi, NN * MM, tid);
  async_copy(gn, Gn, NN * MM, tid);
  async_wait();
  __syncthreads();

  // ---- GEMM1: M=64(h), N=32(j1), K=32(j2): 4x2x2 = 16 wave-jobs
  for (int job = wave; job < 16; job += 8) {
    int ri = job & 1, nt = (job >> 1) & 1, mt = job >> 2;
    v8f acc = V8F_ZERO;
    for (int k0 = 0; k0 < MM; k0 += 4) {
      v2f a1 = load_frag(gr + (mt * 16) * MM + k0, MM);
      v2f b1 = load_frag((ri ? t1ti : t1tr) + (nt * 16) * MM + k0, MM);
      acc = wmma4(a1, b1, acc);
      v2f a2 = load_frag((ri ? gi : gn) + (mt * 16) * MM + k0, MM);
      v2f b2 = load_frag((ri ? t1tr : t1ti) + (nt * 16) * MM + k0, MM);
      acc = wmma4(a2, b2, acc);
    }
    float* usel = ri ? ui : ur;
    int n = nt * 16 + (lane & 15);
    int mb = mt * 16 + 8 * (lane >> 4);
#pragma unroll
    for (int r = 0; r < 8; ++r) usel[(mb + r) * MM + n] = acc[r];
  }
  __syncthreads();

  // ---- GEMM2: M=64(h), N=64(w), K=32(j1): 4x4x2 = 32 wave-jobs
  for (int job = wave; job < 32; job += 8) {
    int ri = job & 1, nt = (job >> 1) & 3, mt = job >> 3;
    v8f acc = V8F_ZERO;
    for (int k0 = 0; k0 < MM; k0 += 4) {
      // out_r = Ur*Gr + Ui*(-Gi) ; out_i = Ur*Gi + Ui*Gr
      v2f a1 = load_frag(ur + (mt * 16) * MM + k0, MM);
      v2f b1 = load_frag((ri ? gi : gr) + (nt * 16) * MM + k0, MM);
      acc = wmma4(a1, b1, acc);
      v2f a2 = load_frag(ui + (mt * 16) * MM + k0, MM);
      v2f b2 = load_frag((ri ? gr : gn) + (nt * 16) * MM + k0, MM);
      acc = wmma4(a2, b2, acc);
    }
    float* dst = out + (size_t)slab * (NN * NN) * 2 + ri;
    int w = nt * 16 + (lane & 15);
    int hb = mt * 16 + 8 * (lane >> 4);
#pragma unroll
    for (int r = 0; r < 8; ++r)
      dst[(size_t)((hb + r) * NN + w) * 2] = acc[r];
  }
}

// ---------------------------------------------------------------------------
// Host-side orchestration (graph-capture safe: kernels + async memset only).
// ---------------------------------------------------------------------------
extern "C" void kernel_launch(void* const* d_in, const int* in_sizes, int n_in,
                              void* d_out, int out_size, void* d_ws, size_t ws_size,
                              hipStream_t stream) {
  (void)in_sizes; (void)n_in; (void)out_size; (void)ws_size;
  const float* x  = (const float*)d_in[0];   // [4,32,64,64,64]
  const float* Wr = (const float*)d_in[1];   // [32,32,16,9]
  const float* Wi = (const float*)d_in[2];
  float* outp = (float*)d_out;               // complex64 interleaved

  float* ws = (float*)d_ws;
  size_t o = 0;
  float* Fr  = ws + o; o += MM * NN;
  float* Fi  = ws + o; o += MM * NN;
  float* Fin = ws + o; o += MM * NN;
  float* Gr  = ws + o; o += NN * MM;
  float* Gi  = ws + o; o += NN * MM;
  float* Gn  = ws + o; o += NN * MM;
  float* Y   = ws + o; o += (size_t)NL * VBLK;
  int*   shell  = (int*)(ws + o); o += VBLK;
  int*   counts = (int*)(ws + o); o += 64;
  float* invcnt = ws + o; o += 64;
  float* cfr = ws + o; o += (size_t)BC * NSL;
  float* cfi = ws + o; o += (size_t)BC * NSL;
  float* c2r = ws + o; o += (size_t)BC * NSL;
  float* c2i = ws + o; o += (size_t)BC * NSL;
  // bufA: s2 (fwd) then reused as t1 (inv): [128][<=64][1024] x {r,i}
  float* bufAr = ws + o; o += (size_t)BC * 64 * 1024;
  float* bufAi = ws + o; o += (size_t)BC * 64 * 1024;
  // bufB: Xb (fwd) then reused as Xrec^T (inv): [128][32768] x {r,i}
  float* bufBr = ws + o; o += (size_t)BC * VBLK;
  float* bufBi = ws + o; o += (size_t)BC * VBLK;

  // --- tables (cheap, rebuilt deterministically every call)
  hipMemsetAsync(counts, 0, 64 * sizeof(int), stream);
  init_dft_kernel<<<8, 256, 0, stream>>>(Fr, Fi, Fin, Gr, Gi, Gn);
  init_basis_kernel<<<VBLK / 256, 256, 0, stream>>>(Y, shell, counts);
  init_counts_kernel<<<1, 32, 0, stream>>>(counts, invcnt);

  // --- forward: W,H axes fused per slab; then D axis as batched cGEMM
  fwd_stage12_kernel<<<BC * NN, 256, 0, stream>>>(x, Fr, Fi, Fin, bufAr, bufAi);
  cgemm_stage_kernel<<<dim3(16, BC), 256, 0, stream>>>(
      bufAr, bufAi, Fr, Fi, Fin, bufBr, bufBi, MM, NN, 0);

  // --- harmonic projection, channel mix, inverse projection
  project_kernel<<<BC, 256, 0, stream>>>(bufBr, bufBi, Y, shell, invcnt, cfr, cfi);
  mix_kernel<<<(4 * 32 * NSL + 255) / 256, 256, 0, stream>>>(cfr, cfi, Wr, Wi, c2r, c2i);
  unproject_kernel<<<dim3(VBLK / 256, BC), 256, 0, stream>>>(
      c2r, c2i, Y, shell, bufBr, bufBi);        // overwrites Xb (dead) as Xrec^T

  // --- inverse: D axis as batched cGEMM (nswap=1); then H,W axes fused
  cgemm_stage_kernel<<<dim3(16, BC), 256, 0, stream>>>(
      bufBr, bufBi, Gr, Gi, Gn, bufAr, bufAi, NN, MM, 1);  // reuses bufA as t1
  inv_stage23_kernel<<<BC * NN, 256, 0, stream>>>(bufAr, bufAi, Gr, Gi, Gn, outp);
}